// Transformer_decoder_net_7009386627330
// MI455X (gfx1250) — compile-verified
//
#include <hip/hip_runtime.h>
#include <hip/hip_bf16.h>

// MI455X / gfx1250, wave32. One workgroup (256 threads = 8 waves) per batch row.
// Memory-bound kernel: 1.31 GB of K/V streamed once -> ~56us floor @ 23.3 TB/s.

typedef __attribute__((ext_vector_type(2))) float v2f;
typedef __attribute__((ext_vector_type(8))) float v8f;

#define NN    1000   // nodes
#define DEMB  128    // embedding dim
#define NH    8      // heads (layers 0,1)
#define TOTD  384    // K_att/V_att row stride (3 layers * 128)
#define NCH   63     // ceil(1000/16) node tiles

__global__ __launch_bounds__(256)
void attn_decoder_kernel(const float* __restrict__ query,
                         const float* __restrict__ Katt,
                         const float* __restrict__ Vatt,
                         const unsigned char* __restrict__ mask,
                         const float* __restrict__ W0w,
                         const float* __restrict__ W0b,
                         const float* __restrict__ Wqw,
                         const float* __restrict__ Wqb,
                         float* __restrict__ out)
{
    const int b    = blockIdx.x;
    const int tid  = threadIdx.x;
    const int lane = tid & 31;
    const int wv   = tid >> 5;          // wave id 0..7
    const int l15  = lane & 15;
    const int kkb  = (lane >> 4) << 1;  // K-sub-offset: lanes 0-15 -> {0,1}, 16-31 -> {2,3}

    __shared__ __align__(16) float qs[DEMB];        // current query
    __shared__ __align__(16) float q2[DEMB];        // attn-out vector / q_final
    __shared__ __align__(16) float lg[NN * NH];     // logits -> softmax weights, [n*8+h]
    __shared__ __align__(16) float red[256 * NH];   // reduction scratch
    __shared__ __align__(16) float outacc[8 * DEMB];
    __shared__ float m8[NH], is8[NH];

    const float* Kb = Katt + (size_t)b * NN * TOTD;
    const float* Vb = Vatt + (size_t)b * NN * TOTD;
    const unsigned char* mb = mask + (size_t)b * NN;

    if (tid < DEMB) qs[tid] = query[(size_t)b * DEMB + tid];
    __syncthreads();

    // ---------------- layers 0 and 1 (8 heads) ----------------
    for (int l = 0; l < 2; ++l) {
        const float* Kl = Kb + l * DEMB;

        // logits[h][n] via V_WMMA_F32_16X16X4_F32, block-diagonal A over heads.
        for (int ch = wv; ch < NCH; ch += 8) {            // uniform per wave -> EXEC all 1s
            int n0   = ch << 4;
            int nrow = n0 + l15; if (nrow > NN - 1) nrow = NN - 1;  // clamp pad rows
            const float* kp = Kl + (size_t)nrow * TOTD + kkb;
            v8f acc = {};
            #pragma unroll
            for (int c = 0; c < 32; ++c) {                // K = 128 in chunks of 4
                v2f bv = *(const v2f*)(kp + 4 * c);       // B[kk, n] = K[n, 4c+kk]
                v2f qa = *(const v2f*)(qs + 4 * c + kkb); // q[4c+kk]
                bool act = (l15 == (c >> 2));             // head owning this K-chunk
                v2f av; av.x = act ? qa.x : 0.0f; av.y = act ? qa.y : 0.0f;
                acc = __builtin_amdgcn_wmma_f32_16x16x4_f32(
                        false, av, false, bv, (short)0, acc, false, false);
            }
            if (lane < 16 && (n0 + lane) < NN) {          // D rows 0..7 live in lanes 0-15
                int n = n0 + lane;
                bool mk = mb[n] != 0;
                float4 lo, hi;
                lo.x = acc[0] * 0.25f; lo.y = acc[1] * 0.25f;   // 1/sqrt(16)
                lo.z = acc[2] * 0.25f; lo.w = acc[3] * 0.25f;
                hi.x = acc[4] * 0.25f; hi.y = acc[5] * 0.25f;
                hi.z = acc[6] * 0.25f; hi.w = acc[7] * 0.25f;
                if (mk) {
                    lo.x = lo.y = lo.z = lo.w = -1.0e9f;
                    hi.x = hi.y = hi.z = hi.w = -1.0e9f;
                }
                *(float4*)(lg + n * 8)     = lo;
                *(float4*)(lg + n * 8 + 4) = hi;
            }
        }
        __syncthreads();

        // per-head max
        float mx[NH];
        #pragma unroll
        for (int h = 0; h < NH; ++h) mx[h] = -3.0e38f;
        for (int n = tid; n < NN; n += 256) {
            #pragma unroll
            for (int h = 0; h < NH; ++h) mx[h] = fmaxf(mx[h], lg[n * 8 + h]);
        }
        #pragma unroll
        for (int h = 0; h < NH; ++h) red[tid * 8 + h] = mx[h];
        __syncthreads();
        for (int s = 128; s > 0; s >>= 1) {
            if (tid < s) {
                #pragma unroll
                for (int h = 0; h < NH; ++h)
                    red[tid * 8 + h] = fmaxf(red[tid * 8 + h], red[(tid + s) * 8 + h]);
            }
            __syncthreads();
        }
        if (tid < NH) m8[tid] = red[tid];
        __syncthreads();

        // per-head sum of exp
        float sm[NH];
        #pragma unroll
        for (int h = 0; h < NH; ++h) sm[h] = 0.0f;
        for (int n = tid; n < NN; n += 256) {
            #pragma unroll
            for (int h = 0; h < NH; ++h) sm[h] += __expf(lg[n * 8 + h] - m8[h]);
        }
        #pragma unroll
        for (int h = 0; h < NH; ++h) red[tid * 8 + h] = sm[h];
        __syncthreads();
        for (int s = 128; s > 0; s >>= 1) {
            if (tid < s) {
                #pragma unroll
                for (int h = 0; h < NH; ++h)
                    red[tid * 8 + h] += red[(tid + s) * 8 + h];
            }
            __syncthreads();
        }
        if (tid < NH) is8[tid] = 1.0f / red[tid];
        __syncthreads();

        // normalize logits -> weights in place
        for (int idx = tid; idx < NN * NH; idx += 256)
            lg[idx] = __expf(lg[idx] - m8[idx & 7]) * is8[idx & 7];
        __syncthreads();

        // out[h,d] = sum_n w[h,n] * V[n,h,d]; wave w handles n = w mod 8,
        // lane covers dims 4*lane..4*lane+3 (head = lane>>2); 512B coalesced row reads.
        const float* Vl = Vb + l * DEMB;
        float4 a4; a4.x = a4.y = a4.z = a4.w = 0.0f;
        const int hh = lane >> 2;
        for (int n = wv; n < NN; n += 8) {
            float wgt = lg[n * 8 + hh];
            float4 v4 = *(const float4*)(Vl + (size_t)n * TOTD + 4 * lane);
            a4.x += wgt * v4.x; a4.y += wgt * v4.y;
            a4.z += wgt * v4.z; a4.w += wgt * v4.w;
        }
        *(float4*)(outacc + wv * DEMB + 4 * lane) = a4;
        __syncthreads();
        if (tid < DEMB) {
            float s = 0.0f;
            #pragma unroll
            for (int w = 0; w < 8; ++w) s += outacc[w * DEMB + tid];
            q2[tid] = s;
        }
        __syncthreads();

        // qs = W0 * q2 + b0   (out @ W0^T + b); W0 row j dotted with out
        {
            int j = tid & 127, hf = tid >> 7;
            const float* Wr = W0w + (size_t)j * DEMB + hf * 64;
            const float* xr = q2 + hf * 64;
            float s = hf ? 0.0f : W0b[j];
            #pragma unroll 8
            for (int i = 0; i < 64; ++i) s += Wr[i] * xr[i];
            red[tid] = s;
        }
        __syncthreads();
        if (tid < DEMB) qs[tid] = red[tid] + red[tid + 128];
        __syncthreads();
    }

    // ---------------- layer 2: q_final = Wq*qs + bq, single head, clip ----------------
    {
        int j = tid & 127, hf = tid >> 7;
        const float* Wr = Wqw + (size_t)j * DEMB + hf * 64;
        const float* xr = qs + hf * 64;
        float s = hf ? 0.0f : Wqb[j];
        #pragma unroll 8
        for (int i = 0; i < 64; ++i) s += Wr[i] * xr[i];
        red[tid] = s;
    }
    __syncthreads();
    if (tid < DEMB) q2[tid] = red[tid] + red[tid + 128];
    __syncthreads();

    const float* K2 = Kb + 2 * DEMB;
    for (int ch = wv; ch < NCH; ch += 8) {
        int n0   = ch << 4;
        int nrow = n0 + l15; if (nrow > NN - 1) nrow = NN - 1;
        const float* kp = K2 + (size_t)nrow * TOTD + kkb;
        v8f acc = {};
        #pragma unroll
        for (int c = 0; c < 32; ++c) {
            v2f bv = *(const v2f*)(kp + 4 * c);
            v2f qa = *(const v2f*)(q2 + 4 * c + kkb);
            bool act = (l15 == 0);                    // only row M=0 valid (1 head)
            v2f av; av.x = act ? qa.x : 0.0f; av.y = act ? qa.y : 0.0f;
            acc = __builtin_amdgcn_wmma_f32_16x16x4_f32(
                    false, av, false, bv, (short)0, acc, false, false);
        }
        if (lane < 16 && (n0 + lane) < NN) {
            int n = n0 + lane;
            float lgt = acc[0] * 0.08838834764831845f; // 1/sqrt(128)
            lgt = 10.0f * tanhf(lgt);                  // clip
            if (mb[n]) lgt = -1.0e9f;
            lg[n] = lgt;
        }
    }
    __syncthreads();

    // single-head softmax -> output weights
    float mx = -3.0e38f;
    for (int n = tid; n < NN; n += 256) mx = fmaxf(mx, lg[n]);
    red[tid] = mx;
    __syncthreads();
    for (int s = 128; s > 0; s >>= 1) {
        if (tid < s) red[tid] = fmaxf(red[tid], red[tid + s]);
        __syncthreads();
    }
    if (tid == 0) m8[0] = red[0];
    __syncthreads();
    float sm = 0.0f;
    for (int n = tid; n < NN; n += 256) sm += __expf(lg[n] - m8[0]);
    red[tid] = sm;
    __syncthreads();
    for (int s = 128; s > 0; s >>= 1) {
        if (tid < s) red[tid] += red[tid + s];
        __syncthreads();
    }
    if (tid == 0) is8[0] = 1.0f / red[0];
    __syncthreads();

    float* op = out + (size_t)b * NN;
    for (int n = tid; n < NN; n += 256)
        op[n] = __expf(lg[n] - m8[0]) * is8[0];
}

extern "C" void kernel_launch(void* const* d_in, const int* in_sizes, int n_in,
                              void* d_out, int out_size, void* d_ws, size_t ws_size,
                              hipStream_t stream) {
    const float* query = (const float*)d_in[0];
    const float* Katt  = (const float*)d_in[1];
    const float* Vatt  = (const float*)d_in[2];
    const unsigned char* mask = (const unsigned char*)d_in[3]; // numpy bool: 1 byte/elem
    const float* W0w = (const float*)d_in[4];
    const float* W0b = (const float*)d_in[5];
    const float* Wqw = (const float*)d_in[6];
    const float* Wqb = (const float*)d_in[7];
    float* out = (float*)d_out;

    int bsz = in_sizes[0] / DEMB;   // 512
    attn_decoder_kernel<<<dim3(bsz), dim3(256), 0, stream>>>(
        query, Katt, Vatt, mask, W0w, W0b, Wqw, Wqb, out);
}